// SwitchRouter_7713761264023
// MI455X (gfx1250) — compile-verified
//
#include <hip/hip_runtime.h>
#include <cmath>

// ---------------- problem constants (from reference) ----------------
constexpr int kB = 16;
constexpr int kT = 2048;
constexpr int kC = 896;
constexpr int kE = 4;
constexpr int kN = kB * kT;          // 32768 tokens
constexpr int kCAP = kN / kE;        // 8192 (capacity_factor 1.0)
constexpr float kAuxW = 0.01f;
constexpr float kZW = 0.001f;

// ---------------- tiling ----------------
constexpr int TPB   = 128;   // tokens per block (8 waves x 16 tokens)
constexpr int KC    = 32;    // K-chunk staged in LDS by the TDM
constexpr int ASTR  = 36;    // LDS row stride: 32 data + 4 pad DWORDs (TDM pad_enable)
constexpr int WSTR  = 904;   // padded LDS row stride for W
constexpr int NBUCK = 4096;  // selection histogram buckets (key bits [24:13])
constexpr int CANDMAX = 8192;

typedef __attribute__((ext_vector_type(2))) float    v2f;
typedef __attribute__((ext_vector_type(4))) float    v4f;
typedef __attribute__((ext_vector_type(8))) float    v8f;
typedef __attribute__((ext_vector_type(4))) unsigned v4u;
typedef __attribute__((ext_vector_type(4))) int      v4i;
typedef __attribute__((ext_vector_type(8))) int      v8i;

struct Meta {
    float    psum[4];     // sum of router_probs per expert (for P)
    float    zsum;        // sum of lse^2 (for z-loss)
    unsigned candcnt[4];  // tokens landing in the threshold bucket
    int      selall[4];   // expert not capped -> select everything assigned
    int      bstar[4];    // threshold bucket
    int      r[4];        // slots remaining inside threshold bucket
    int      ne[4];       // tokens assigned per expert
    int      selected[4]; // min(ne, CAP)
};
constexpr size_t HIST_BYTES = (size_t)kE * NBUCK * sizeof(unsigned);
constexpr size_t META_PAD   = 128;

// =====================================================================
// TDM: 2D tile load  global[128 rows x 32 f32, row stride 896] -> LDS,
// with pad_enable inserting 4 DWORDs after every 32 stored DWORDs so
// the LDS tile lands with row stride 36 (bank-conflict-free for the
// 16-lane column reads of the WMMA A-fragments).
// Descriptor per CDNA5 ISA.
// =====================================================================
__device__ __forceinline__ void tdm_load_tile(const float* gsrc, unsigned lds_byte_off)
{
    unsigned long long ga = (unsigned long long)(uintptr_t)gsrc;

    v4u g0;
    g0[0] = 1u;                                   // count=1, user mode, no gather
    g0[1] = lds_byte_off;                         // lds_addr
    g0[2] = (unsigned)ga;                         // global_addr[31:0]
    g0[3] = (unsigned)(ga >> 32) | (2u << 30);    // global_addr[56:32] | type=2

    v8i g1;
    g1[0] = (int)((2u << 16)        // data_size = 4 bytes
                | (1u << 20)        // pad_enable
                | (4u << 22)        // pad_interval: 32 DWORDs (enc: 1<<(n+1))
                | (3u << 25));      // pad_amount: 4 DWORDs (enc: n+1)
    g1[1] = (int)((unsigned)KC << 16);   // tensor_dim0[15:0] = 32 (bits 63:48)
    g1[2] = (int)((unsigned)TPB << 16);  // tensor_dim1[15:0] = 128 (bits 95:80)
    g1[3] = (int)((unsigned)KC << 16);   // tile_dim0 = 32 (bits 127:112)
    g1[4] = TPB;                         // tile_dim1 = 128, tile_dim2 = 0
    g1[5] = kC;                          // tensor_dim0_stride = 896 elements
    g1[6] = 0;
    g1[7] = 0;

    v4i gz = {0, 0, 0, 0};               // groups 2/3 unused (<=2D tensor)
#if __clang_major__ >= 23
    v8i gz8 = {0, 0, 0, 0, 0, 0, 0, 0};
    __builtin_amdgcn_tensor_load_to_lds(g0, g1, gz, gz, gz8, 0);
#else
    __builtin_amdgcn_tensor_load_to_lds(g0, g1, gz, gz, 0);
#endif
}

// =====================================================================
// Kernel 1: logits via V_WMMA_F32_16X16X4_F32 with TDM double-buffered
//           A tiles; softmax(E=4), argmax, selection key + histogram,
//           loss partial sums.
// =====================================================================
__global__ __launch_bounds__(256)
void router_wmma_kernel(const float* __restrict__ xg,
                        const float* __restrict__ Wg,
                        unsigned* __restrict__ hist,
                        Meta* __restrict__ meta,
                        unsigned* __restrict__ keys,
                        int* __restrict__ experts)
{
    __shared__ float Alds[2 * TPB * ASTR];   // 36864 B (double-buffered A tile)
    __shared__ float Wl[(kE + 1) * WSTR];    // 18080 B (row 4 = zeros, pads E 4->16)
    __shared__ float acc[8];                 // psum[0..3], zsum

    const int t    = threadIdx.x;
    const int w    = t >> 5;             // wave id (wave32)
    const int lane = t & 31;
    const int hf   = lane >> 4;          // half-wave select (K split per ISA layout)
    const int l15  = lane & 15;
    const int tokbase = blockIdx.x * TPB;

    const unsigned ldsA0 = (unsigned)(unsigned long long)(uintptr_t)(void*)&Alds[0];
    constexpr unsigned BUFB = (unsigned)(TPB * ASTR * sizeof(float));

    if (t < 8) acc[t] = 0.0f;

    // Stage W (4 x 896 f32) into LDS, padded stride; row 4 is all zeros so
    // B-fragment loads are unconditional (no exec divergence in the hot loop).
    for (int i = t; i < kE * kC; i += 256) {
        int e = i / kC, c = i - e * kC;
        Wl[e * WSTR + c] = Wg[i];
    }
    for (int i = t; i < kC; i += 256) Wl[kE * WSTR + i] = 0.0f;

    // Per-lane B row base: experts 0..3 real, lanes 4..15 -> zero row (broadcast).
    const float* bbase = &Wl[(l15 < kE ? l15 : kE) * WSTR];
    const float* abase = &Alds[(w * 16 + l15) * ASTR + 2 * hf];

    v8f cfrag = {0.f, 0.f, 0.f, 0.f, 0.f, 0.f, 0.f, 0.f};

    // Prologue: wave 0 kicks the first TDM tile into buffer 0.
    if (w == 0)
        tdm_load_tile(&xg[(size_t)tokbase * kC], ldsA0);

    for (int kc = 0; kc < kC; kc += KC) {
        const int cur = (kc / KC) & 1;
        if (w == 0) __builtin_amdgcn_s_wait_tensorcnt(0);
        __syncthreads();  // tile `cur` ready; everyone done reading tile `cur^1`

        // Overlap: DMA next chunk into the other buffer while we do WMMAs.
        if (w == 0 && kc + KC < kC)
            tdm_load_tile(&xg[(size_t)tokbase * kC + kc + KC], ldsA0 + (cur ^ 1) * BUFB);
        // Software prefetch of chunk kc+2*KC cachelines (128 rows x 128B).
        if (w == 1 && kc + 2 * KC < kC) {
            #pragma unroll
            for (int r = 0; r < 4; ++r)
                __builtin_prefetch(&xg[(size_t)(tokbase + lane + 32 * r) * kC + kc + 2 * KC], 0, 3);
        }

        // 8 WMMAs per chunk per wave: D(16x16) += A(16x4) * B(4x16).
        // A layout: lanes 0-15 hold K={0,1}, lanes 16-31 hold K={2,3}; B mirrors.
        const float* a0 = abase + cur * (TPB * ASTR);
        #pragma unroll
        for (int kk = 0; kk < KC; kk += 4) {
            v2f a = *(const v2f*)&a0[kk];
            v2f b = *(const v2f*)&bbase[kc + kk + 2 * hf];
            cfrag = __builtin_amdgcn_wmma_f32_16x16x4_f32(
                false, a, false, b, (short)0, cfrag, false, false);
        }
    }
    __syncthreads();

    // Spill C fragment to LDS so each token's 4 logits land in one lane.
    // C/D layout: VGPR j -> M = j + 8*hf, N = lane&15.
    float* scratch = &Alds[w * 272];     // 16x17 per wave, disjoint regions
    #pragma unroll
    for (int j = 0; j < 8; ++j)
        scratch[(j + 8 * hf) * 17 + l15] = cfrag[j];
    __syncthreads();

    if (lane < 16) {
        const int token = tokbase + w * 16 + lane;
        float l0 = scratch[lane * 17 + 0];
        float l1 = scratch[lane * 17 + 1];
        float l2 = scratch[lane * 17 + 2];
        float l3 = scratch[lane * 17 + 3];

        float mx = fmaxf(fmaxf(l0, l1), fmaxf(l2, l3));
        float e0 = expf(l0 - mx), e1 = expf(l1 - mx);
        float e2 = expf(l2 - mx), e3 = expf(l3 - mx);
        float s  = e0 + e1 + e2 + e3;
        float inv = 1.0f / s;

        // argmax with first-max tie-break (matches jnp.argmax)
        int be = 0; float bl = l0;
        if (l1 > bl) { bl = l1; be = 1; }
        if (l2 > bl) { bl = l2; be = 2; }
        if (l3 > bl) { bl = l3; be = 3; }
        float pbest = (be == 0 ? e0 : be == 1 ? e1 : be == 2 ? e2 : e3) * inv;

        float lse = mx + logf(s);

        unsigned kbits = __float_as_uint(pbest);  // p>0 => bit order == float order
        keys[token]    = kbits;
        experts[token] = be;
        atomicAdd(&hist[be * NBUCK + ((kbits >> 13) & (NBUCK - 1))], 1u);

        atomicAdd(&acc[0], e0 * inv);
        atomicAdd(&acc[1], e1 * inv);
        atomicAdd(&acc[2], e2 * inv);
        atomicAdd(&acc[3], e3 * inv);
        atomicAdd(&acc[4], lse * lse);
    }
    __syncthreads();
    if (t < 5) {
        float v = acc[t];
        if (t < 4) atomicAdd(&meta->psum[t], v);
        else       atomicAdd(&meta->zsum, v);
    }
}

// =====================================================================
// Kernel 2: per-expert suffix scan of the histogram -> threshold bucket.
// =====================================================================
__global__ __launch_bounds__(256)
void scan_kernel(const unsigned* __restrict__ hist, Meta* __restrict__ meta)
{
    __shared__ unsigned csum[256];
    const int e = blockIdx.x;
    const int t = threadIdx.x;
    const unsigned* h = hist + e * NBUCK;
    constexpr int BPT = NBUCK / 256;     // 16 bins per thread

    unsigned s = 0;
    for (int i = 0; i < BPT; ++i) s += h[t * BPT + i];
    csum[t] = s;
    __syncthreads();

    if (t == 0) {
        unsigned total = 0;
        for (int i = 0; i < 256; ++i) total += csum[i];
        meta->ne[e] = (int)total;
        if ((int)total <= kCAP) {
            meta->selall[e]   = 1;
            meta->bstar[e]    = -1;
            meta->r[e]        = 0;
            meta->selected[e] = (int)total;
        } else {
            meta->selall[e]   = 0;
            meta->selected[e] = kCAP;
            unsigned cum = 0; int bstar = 0; unsigned cab = 0;
            for (int c = 255; c >= 0; --c) {
                if (cum + csum[c] >= (unsigned)kCAP) {
                    for (int b = c * BPT + BPT - 1; b >= c * BPT; --b) {
                        if (cum + h[b] >= (unsigned)kCAP) { bstar = b; cab = cum; break; }
                        cum += h[b];
                    }
                    break;
                }
                cum += csum[c];
            }
            meta->bstar[e] = bstar;
            meta->r[e]     = kCAP - (int)cab;
        }
    }
}

// =====================================================================
// Kernel 3: per-token decision + bulk output write; threshold-bucket
//           tokens become candidates for exact ranking.
// =====================================================================
__global__ __launch_bounds__(256)
void assign_kernel(const unsigned* __restrict__ keys,
                   const int* __restrict__ experts,
                   Meta* __restrict__ meta,
                   int* __restrict__ cand,
                   float* __restrict__ out)
{
    const int i = blockIdx.x * 256 + threadIdx.x;
    if (i >= kN) return;
    const int e = experts[i];
    const unsigned kb = keys[i];
    const int b = (kb >> 13) & (NBUCK - 1);

    float val = 0.0f;
    if (meta->selall[e]) {
        val = 1.0f;
    } else {
        const int bs = meta->bstar[e];
        if (b > bs) {
            val = 1.0f;
        } else if (b == bs) {
            unsigned pos = atomicAdd(&meta->candcnt[e], 1u);
            if (pos < (unsigned)CANDMAX) cand[e * CANDMAX + (int)pos] = i;
        }
    }
    v4f g = {e == 0 ? val : 0.f, e == 1 ? val : 0.f,
             e == 2 ? val : 0.f, e == 3 ? val : 0.f};
    *(v4f*)&out[(size_t)i * 4]                  = g;   // gates
    *(v4f*)&out[(size_t)kN * 4 + (size_t)i * 4] = g;   // dispatch_mask
}

// =====================================================================
// Kernel 4: exact rank (key desc, index asc — top_k tie-break) among
//           threshold-bucket candidates; top r get selected.
// =====================================================================
__global__ __launch_bounds__(256)
void subsel_kernel(const unsigned* __restrict__ keys,
                   const int* __restrict__ cand,
                   const Meta* __restrict__ meta,
                   float* __restrict__ out)
{
    const int e = blockIdx.x;
    if (meta->selall[e]) return;
    int m = (int)meta->candcnt[e];
    if (m > CANDMAX) m = CANDMAX;
    const int r = meta->r[e];
    const int* ce = cand + e * CANDMAX;

    for (int ci = threadIdx.x; ci < m; ci += blockDim.x) {
        const int i = ce[ci];
        const unsigned ki = keys[i];
        int rank = 0;
        for (int j = 0; j < m; ++j) {
            const int ij = ce[j];
            const unsigned kj = keys[ij];
            rank += (int)((kj > ki) || (kj == ki && ij < i));
        }
        if (rank < r) {
            out[(size_t)i * 4 + e] = 1.0f;
            out[(size_t)kN * 4 + (size_t)i * 4 + e] = 1.0f;
        }
    }
}

// =====================================================================
// Kernel 5: scalar loss = aux + z.
// =====================================================================
__global__ void loss_kernel(const Meta* __restrict__ meta, float* __restrict__ out)
{
    if (threadIdx.x == 0 && blockIdx.x == 0) {
        float aux = 0.0f;
        for (int e = 0; e < 4; ++e)
            aux += ((float)meta->selected[e] / (float)kN) * (meta->psum[e] / (float)kN);
        aux *= kAuxW * (float)kE;
        float z = kZW * meta->zsum / (float)kN;
        out[(size_t)2 * kN * 4] = aux + z;
    }
}

// =====================================================================
extern "C" void kernel_launch(void* const* d_in, const int* in_sizes, int n_in,
                              void* d_out, int out_size, void* d_ws, size_t ws_size,
                              hipStream_t stream)
{
    const float* x = (const float*)d_in[0];   // [16,2048,896] f32
    const float* W = (const float*)d_in[1];   // [4,896] f32
    float* out = (float*)d_out;               // gates | dispatch | loss

    char* ws = (char*)d_ws;
    unsigned* hist    = (unsigned*)ws;
    Meta*     meta    = (Meta*)(ws + HIST_BYTES);
    unsigned* keys    = (unsigned*)(ws + HIST_BYTES + META_PAD);
    int*      experts = (int*)(ws + HIST_BYTES + META_PAD + (size_t)kN * 4);
    int*      cand    = (int*)(ws + HIST_BYTES + META_PAD + (size_t)kN * 8);

    // Idempotent per launch: zero histogram + meta (graph-capturable).
    hipMemsetAsync(d_ws, 0, HIST_BYTES + sizeof(Meta), stream);

    router_wmma_kernel<<<kN / TPB, 256, 0, stream>>>(x, W, hist, meta, keys, experts);
    scan_kernel<<<kE, 256, 0, stream>>>(hist, meta);
    assign_kernel<<<kN / 256, 256, 0, stream>>>(keys, experts, meta, cand, out);
    subsel_kernel<<<kE, 256, 0, stream>>>(keys, cand, meta, out);
    loss_kernel<<<1, 1, 0, stream>>>(meta, out);
}